// CustomRNN_40810779246611
// MI455X (gfx1250) — compile-verified
//
#include <hip/hip_runtime.h>
#include <hip/hip_bf16.h>
#include <stdint.h>

#define EXT   4080          // NX + 2*PML
#define NXV   4000
#define DT2   2.5e-7f       // DT*DT, DT = 0.0005
#define INV_DX2 0.01f       // 1/(DX*DX), DX = 10

// CDNA5 async global->LDS copy path (ASYNCcnt-tracked). Guarded so the file
// still compiles if this toolchain lacks the builtins; fallback is a plain
// vector load+LDS store.
#if __has_builtin(__builtin_amdgcn_global_load_async_to_lds_b128) && \
    __has_builtin(__builtin_amdgcn_s_wait_asynccnt)
#define ASYNC_LDS 1
#else
#define ASYNC_LDS 0
#endif

// The builtin's real signature (from the compiler diagnostic) takes non-const
// v4i pointers: (v4i addrspace(1)*, v4i addrspace(3)*, imm offset, imm cpol).
typedef int v4i __attribute__((ext_vector_type(4)));
typedef v4i __attribute__((address_space(1)))* gas1_v4i;
typedef v4i __attribute__((address_space(3)))* gas3_v4i;

__device__ __forceinline__ void cp16_to_lds(const float* g, float* l) {
#if ASYNC_LDS
    __builtin_amdgcn_global_load_async_to_lds_b128((gas1_v4i)g, (gas3_v4i)l,
                                                   /*imm offset*/0, /*cpol*/0);
#else
    float4 t = *(const float4*)g;
    *(float4*)l = t;
#endif
}

// ---------------------------------------------------------------------------
// Main stencil: each block computes a 32(x) x 16(y) tile of the interior
// region rows/cols [8, 4072). LDS region per array: 18 rows x 40 floats,
// covering global rows [i0-1, i0+16] and cols [j0-4, j0+35] (16B aligned).
// ---------------------------------------------------------------------------
__global__ __launch_bounds__(256) void wave_step_kernel(
    const float* __restrict__ p1g,
    const float* __restrict__ p2g,
    const float* __restrict__ pmlg,
    const float* __restrict__ varr,
    float* __restrict__ pout)
{
    __shared__ float sP2[18 * 40];
    __shared__ float sPM[18 * 40];

    const int tid = threadIdx.x;
    const int j0  = 8 + 32 * blockIdx.x;     // first output col of tile
    const int i0  = 8 + 16 * blockIdx.y;     // first output row of tile
    const int r0  = i0 - 1;                  // LDS region origin (row)
    const int c0  = j0 - 4;                  // LDS region origin (col), 16B aligned

    const int lx  = tid & 31;                // output col within tile
    const int lyB = tid >> 5;                // base output row (0..7)

    // Prefetch the streamed p1 line for this thread's first row (L2 warm-up).
    __builtin_prefetch(p1g + (size_t)(i0 + lyB) * EXT + j0 + lx, 0, 0);

    // 18 rows * 10 float4 per row = 180 b128 copies per array.
    if (tid < 180) {
        const int r  = tid / 10;
        const int c4 = (tid % 10) * 4;
        const size_t goff = (size_t)(r0 + r) * EXT + (size_t)(c0 + c4);
        const int    loff = r * 40 + c4;
        cp16_to_lds(p2g  + goff, &sP2[loff]);
        cp16_to_lds(pmlg + goff, &sPM[loff]);
    }
#if ASYNC_LDS
    __builtin_amdgcn_s_wait_asynccnt(0);     // s_wait_asynccnt 0
#endif
    __syncthreads();

#pragma unroll
    for (int k = 0; k < 2; ++k) {
        const int ly = lyB + 8 * k;          // output row within tile (0..15)
        const int i  = i0 + ly;
        const int j  = j0 + lx;
        const int lr = ly + 1;               // LDS row
        const int lc = lx + 4;               // LDS col

        const float pmc = sPM[lr * 40 + lc];
        const float cc  = sP2[lr * 40 + lc]       * pmc;
        const float xm  = sP2[lr * 40 + lc - 1]   * sPM[lr * 40 + lc - 1];
        const float xp  = sP2[lr * 40 + lc + 1]   * sPM[lr * 40 + lc + 1];
        const float ym  = sP2[(lr - 1) * 40 + lc] * sPM[(lr - 1) * 40 + lc];
        const float yp  = sP2[(lr + 1) * 40 + lc] * sPM[(lr + 1) * 40 + lc];

        const float lap = (xm + xp + ym + yp - 4.0f * cc) * INV_DX2;

        int ci = i - 40; ci = ci < 0 ? 0 : (ci > NXV - 1 ? NXV - 1 : ci);
        int cj = j - 40; cj = cj < 0 ? 0 : (cj > NXV - 1 ? NXV - 1 : cj);
        const float v     = varr[(size_t)ci * NXV + cj];
        const float alpha = v * v * DT2;

        const float p1c = p1g[(size_t)i * EXT + j] * pmc;

        pout[(size_t)i * EXT + j] = 2.0f * cc - p1c + alpha * lap;
    }
}

// ---------------------------------------------------------------------------
// Zero the 8-wide border frame (4080^2 - 4064^2 = 130304 cells).
// ---------------------------------------------------------------------------
__global__ __launch_bounds__(256) void border_zero_kernel(float* __restrict__ p)
{
    int idx = blockIdx.x * blockDim.x + threadIdx.x;
    const int TOP  = 8 * EXT;        // 32640
    const int SIDE = 4064 * 8;       // 32512
    if (idx < TOP) { p[idx] = 0.0f; return; }
    idx -= TOP;
    if (idx < TOP) { p[(size_t)4072 * EXT + idx] = 0.0f; return; }
    idx -= TOP;
    if (idx < SIDE) {
        const int r = idx >> 3, c = idx & 7;
        p[(size_t)(8 + r) * EXT + c] = 0.0f; return;
    }
    idx -= SIDE;
    if (idx < SIDE) {
        const int r = idx >> 3, c = idx & 7;
        p[(size_t)(8 + r) * EXT + 4072 + c] = 0.0f;
    }
}

// ---------------------------------------------------------------------------
// Tail: inject point source at p[40][60], extract res[:,50] = p[40+i][90].
// (col 60 != col 90, so the source add cannot race the column reads.)
// ---------------------------------------------------------------------------
__global__ __launch_bounds__(256) void tail_kernel(
    const float* __restrict__ sf, const int* __restrict__ t,
    float* __restrict__ p, float* __restrict__ outcol)
{
    const int i = blockIdx.x * blockDim.x + threadIdx.x;
    if (i == 0) {
        p[(size_t)40 * EXT + 60] += sf[t[0]] * DT2;
    }
    if (i < NXV) {
        outcol[i] = p[(size_t)(40 + i) * EXT + 90];
    }
}

extern "C" void kernel_launch(void* const* d_in, const int* in_sizes, int n_in,
                              void* d_out, int out_size, void* d_ws, size_t ws_size,
                              hipStream_t stream)
{
    (void)in_sizes; (void)n_in; (void)out_size; (void)d_ws; (void)ws_size;

    const float* p1   = (const float*)d_in[0];
    const float* p2   = (const float*)d_in[1];
    const float* pml  = (const float*)d_in[2];
    const float* varr = (const float*)d_in[3];
    const float* sf   = (const float*)d_in[4];
    const int*   t    = (const int*)d_in[5];

    float* p   = (float*)d_out;
    float* col = p + (size_t)EXT * EXT;

    border_zero_kernel<<<(130304 + 255) / 256, 256, 0, stream>>>(p);

    dim3 grid(127, 254);   // 4064/32 x 4064/16
    wave_step_kernel<<<grid, 256, 0, stream>>>(p1, p2, pml, varr, p);

    tail_kernel<<<16, 256, 0, stream>>>(sf, t, p, col);
}